// ASAGE_38912403702070
// MI455X (gfx1250) — compile-verified
//
#include <hip/hip_runtime.h>

#define N_USER 100000
#define M_ITEM 50000
#define BATCH  2048
#define SNB    20

typedef float v2f __attribute__((ext_vector_type(2)));
typedef float v8f __attribute__((ext_vector_type(8)));

__device__ __forceinline__ v8f wmma4(v2f a, v2f b, v8f c) {
  // V_WMMA_F32_16X16X4_F32 : exact fp32 D = A(16x4) * B(4x16) + C
  return __builtin_amdgcn_wmma_f32_16x16x4_f32(false, a, false, b, (short)0, c,
                                               false, false);
}

// ---------------------------------------------------------------------------
// Generic gather + group-mean over 64-wide rows. One wave per group.
// idx==nullptr -> rows are g*gsize + r (contiguous groups).
// ---------------------------------------------------------------------------
__global__ __launch_bounds__(256) void mean_gather_kernel(
    const float* __restrict__ src, const int* __restrict__ idx,
    float* __restrict__ out, int ngroups, int gsize, float inv)
{
  int wid = blockIdx.x * (blockDim.x >> 5) + (threadIdx.x >> 5);
  if (wid >= ngroups) return;
  int lane = threadIdx.x & 31;
  float s0 = 0.f, s1 = 0.f;
  for (int r = 0; r < gsize; ++r) {
    int row = idx ? idx[(size_t)wid * gsize + r] : (wid * gsize + r);
    const float* p = src + (size_t)row * 64;
    s0 += p[lane];
    s1 += p[lane + 32];
  }
  out[(size_t)wid * 64 + lane]      = s0 * inv;
  out[(size_t)wid * 64 + lane + 32] = s1 * inv;
}

// ---------------------------------------------------------------------------
// Stage the common 524-float region of 16 node rows into LDS:
//   [id 64 | feat-bag mean 64 | text mean 96 | word300 300]
// ---------------------------------------------------------------------------
__device__ __forceinline__ void fill524(
    float* __restrict__ in, int stride, int row0, int tid, int nth,
    const int* __restrict__ fidx_l,   // LDS [16*10]
    const int* __restrict__ tidx_l,   // LDS [16*24]
    const float* __restrict__ id_emb,
    const float* __restrict__ feat_emb,
    const float* __restrict__ word_emb,
    const float* __restrict__ word300)
{
  // id region [0,64): 16 rows x 16 float4
  for (int i = tid; i < 256; i += nth) {
    int r = i >> 4, c = i & 15;
    const float4 v = ((const float4*)(id_emb + (size_t)(row0 + r) * 64))[c];
    *(float4*)&in[r * stride + c * 4] = v;
  }
  // feature-bag mean region [64,128)
  for (int i = tid; i < 256; i += nth) {
    int r = i >> 4, c = i & 15;
    float ax = 0.f, ay = 0.f, az = 0.f, aw = 0.f;
    for (int j = 0; j < 10; ++j) {
      int id = fidx_l[r * 10 + j];
      const float4 v = ((const float4*)(feat_emb + (size_t)id * 64))[c];
      ax += v.x; ay += v.y; az += v.z; aw += v.w;
    }
    float4 o; o.x = ax * 0.1f; o.y = ay * 0.1f; o.z = az * 0.1f; o.w = aw * 0.1f;
    *(float4*)&in[r * stride + 64 + c * 4] = o;
  }
  // text mean region [128,224): 3 sources x 32 floats, mean over 8 tokens
  for (int i = tid; i < 384; i += nth) {
    int r = i / 24, c = i % 24;
    int s = c >> 3, f = c & 7;
    float ax = 0.f, ay = 0.f, az = 0.f, aw = 0.f;
    for (int t = 0; t < 8; ++t) {
      int id = tidx_l[r * 24 + s * 8 + t];
      const float4 v = ((const float4*)(word_emb + (size_t)id * 32))[f];
      ax += v.x; ay += v.y; az += v.z; aw += v.w;
    }
    float4 o; o.x = ax * 0.125f; o.y = ay * 0.125f; o.z = az * 0.125f; o.w = aw * 0.125f;
    *(float4*)&in[r * stride + 128 + c * 4] = o;
  }
  // word300 region [224,524): 75 float4 per row
  for (int i = tid; i < 16 * 75; i += nth) {
    int r = i / 75, c = i % 75;
    const float4 v = ((const float4*)(word300 + (size_t)(row0 + r) * 300))[c];
    *(float4*)&in[r * stride + 224 + c * 4] = v;
  }
}

// Per-wave WMMA K-loop over one staged LDS tile region against row-major W.
// Wave handles n-columns [wave*32, wave*32+32). Accumulates into acc0/acc1.
__device__ __forceinline__ void wmma_kloop(
    const float* __restrict__ in, int stride, int kmax,
    const float* __restrict__ W, int wrow0,
    int lo, int hi, int n0, v8f& acc0, v8f& acc1)
{
  for (int k0 = 0; k0 < kmax; k0 += 4) {
    const int kb = k0 + 2 * hi;              // fp32 A layout: K pair per lane
    v2f a = *(const v2f*)&in[lo * stride + kb];
    const float* wp = W + (size_t)(wrow0 + kb) * 64;
    v2f b0, b1;
    b0[0] = wp[n0];          b0[1] = wp[64 + n0];
    b1[0] = wp[n0 + 16];     b1[1] = wp[64 + n0 + 16];
    acc0 = wmma4(a, b0, acc0);
    acc1 = wmma4(a, b1, acc1);
  }
}

// Epilogue: bias + numeric@Wnum tail + store (C layout: VGPR j -> M=j+8*hi).
__device__ __forceinline__ void embed_epilogue(
    v8f acc0, v8f acc1, int n0, int lo, int hi, int row0,
    const float* __restrict__ num_l,      // LDS [16*10]
    const float* __restrict__ Wnum, const float* __restrict__ bnum,
    const float* __restrict__ bproj, float* __restrict__ xout)
{
  for (int t = 0; t < 2; ++t) {
    v8f acc = t ? acc1 : acc0;
    const int n = n0 + t * 16;
    float wn[10];
    for (int k = 0; k < 10; ++k) wn[k] = Wnum[k * 64 + n];
    const float bb = bproj[n] + bnum[n];
    for (int j = 0; j < 8; ++j) {
      const int m = j + 8 * hi;
      float v = acc[j] + bb;
      for (int k = 0; k < 10; ++k) v += num_l[m * 10 + k] * wn[k];
      xout[(size_t)(row0 + m) * 64 + n] = v;
    }
  }
}

// ---------------------------------------------------------------------------
// User node embedding: x[row] = concat524 @ Wproj_u + bproj + num @ Wnum + bnum
// 16 rows per block, 2 waves (each owns 2 of the 4 N-tiles).
// ---------------------------------------------------------------------------
__global__ __launch_bounds__(64) void embed_user_kernel(
    const int* __restrict__ ufeat_idx, const int* __restrict__ utext_idx,
    const float* __restrict__ id_emb, const float* __restrict__ feat_emb,
    const float* __restrict__ word_emb, const float* __restrict__ word300,
    const float* __restrict__ numeric,
    const float* __restrict__ Wnum, const float* __restrict__ bnum,
    const float* __restrict__ Wproj, const float* __restrict__ bproj,
    float* __restrict__ xout)
{
  __shared__ float in[16 * 524];
  __shared__ int   fidx_l[160];
  __shared__ int   tidx_l[384];
  __shared__ float num_l[160];
  const int tid  = threadIdx.x;
  const int row0 = blockIdx.x * 16;

  for (int i = tid; i < 160; i += 64) fidx_l[i] = ufeat_idx[(size_t)row0 * 10 + i];
  for (int i = tid; i < 384; i += 64) tidx_l[i] = utext_idx[(size_t)row0 * 24 + i];
  for (int i = tid; i < 160; i += 64) num_l[i]  = numeric[(size_t)row0 * 10 + i];
  __syncthreads();
  fill524(in, 524, row0, tid, 64, fidx_l, tidx_l, id_emb, feat_emb, word_emb, word300);
  __syncthreads();

  const int lane = tid & 31, wave = tid >> 5;
  const int lo = lane & 15, hi = lane >> 4;
  const int n0 = wave * 32 + lo;
  v8f z = {0.f,0.f,0.f,0.f,0.f,0.f,0.f,0.f};
  v8f acc0 = z, acc1 = z;
  wmma_kloop(in, 524, 524, Wproj, 0, lo, hi, n0, acc0, acc1);
  embed_epilogue(acc0, acc1, n0, lo, hi, row0, num_l, Wnum, bnum, bproj, xout);
}

// ---------------------------------------------------------------------------
// Item node embedding (K=1292): two LDS passes (524 common + 768 sent) to
// stay within LDS limits, sharing one WMMA accumulator set.
// ---------------------------------------------------------------------------
__global__ __launch_bounds__(64) void embed_item_kernel(
    const int* __restrict__ ifeat_idx, const int* __restrict__ itext_idx,
    const float* __restrict__ id_emb, const float* __restrict__ feat_emb,
    const float* __restrict__ word_emb, const float* __restrict__ word300,
    const float* __restrict__ sent768, const float* __restrict__ numeric,
    const float* __restrict__ Wnum, const float* __restrict__ bnum,
    const float* __restrict__ Wproj, const float* __restrict__ bproj,
    float* __restrict__ xout)
{
  __shared__ float in[16 * 768];
  __shared__ int   fidx_l[160];
  __shared__ int   tidx_l[384];
  __shared__ float num_l[160];
  const int tid  = threadIdx.x;
  const int row0 = blockIdx.x * 16;

  for (int i = tid; i < 160; i += 64) fidx_l[i] = ifeat_idx[(size_t)row0 * 10 + i];
  for (int i = tid; i < 384; i += 64) tidx_l[i] = itext_idx[(size_t)row0 * 24 + i];
  for (int i = tid; i < 160; i += 64) num_l[i]  = numeric[(size_t)row0 * 10 + i];
  __syncthreads();
  fill524(in, 524, row0, tid, 64, fidx_l, tidx_l, id_emb, feat_emb, word_emb, word300);
  __syncthreads();

  const int lane = tid & 31, wave = tid >> 5;
  const int lo = lane & 15, hi = lane >> 4;
  const int n0 = wave * 32 + lo;
  v8f z = {0.f,0.f,0.f,0.f,0.f,0.f,0.f,0.f};
  v8f acc0 = z, acc1 = z;
  wmma_kloop(in, 524, 524, Wproj, 0, lo, hi, n0, acc0, acc1);
  __syncthreads();                       // done reading pass-1 tile

  // pass 2: sent768 region (K rows 524..1291)
  for (int i = tid; i < 16 * 192; i += 64) {
    int r = i / 192, c = i % 192;
    const float4 v = ((const float4*)(sent768 + (size_t)(row0 + r) * 768))[c];
    *(float4*)&in[r * 768 + c * 4] = v;
  }
  __syncthreads();
  wmma_kloop(in, 768, 768, Wproj, 524, lo, hi, n0, acc0, acc1);
  embed_epilogue(acc0, acc1, n0, lo, hi, row0, num_l, Wnum, bnum, bproj, xout);
}

// ---------------------------------------------------------------------------
// Fused SAGE layer for a 16-row tile (one wave):
//   tmp = nm @ Wv + bv ;  out = [self | tmp] @ Ww + bw ; optional relu
// Uses WMMA for both matmuls; tmp round-trips through LDS to convert the
// C layout to an A fragment layout.
// ---------------------------------------------------------------------------
__global__ __launch_bounds__(32) void sage_kernel(
    const float* __restrict__ selfb, const float* __restrict__ nmean,
    const float* __restrict__ Wv, const float* __restrict__ bv,
    const float* __restrict__ Ww, const float* __restrict__ bw,
    float* __restrict__ out, int do_relu)
{
  __shared__ float sl[16 * 68];
  __shared__ float nl[16 * 68];
  __shared__ float tl[16 * 68];
  const int tid  = threadIdx.x;
  const int row0 = blockIdx.x * 16;

  for (int i = tid; i < 256; i += 32) {
    int r = i >> 4, c = i & 15;
    float4 a = ((const float4*)(selfb + (size_t)(row0 + r) * 64))[c];
    float4 b = ((const float4*)(nmean + (size_t)(row0 + r) * 64))[c];
    *(float4*)&sl[r * 68 + c * 4] = a;
    *(float4*)&nl[r * 68 + c * 4] = b;
  }
  __syncthreads();

  const int lo = tid & 15, hi = tid >> 4;
  v8f z = {0.f,0.f,0.f,0.f,0.f,0.f,0.f,0.f};
  v8f acc[4] = {z, z, z, z};
  for (int k0 = 0; k0 < 64; k0 += 4) {
    const int kb = k0 + 2 * hi;
    v2f a = *(const v2f*)&nl[lo * 68 + kb];
    const float* wp = Wv + (size_t)kb * 64;
#pragma unroll
    for (int nt = 0; nt < 4; ++nt) {
      v2f b; b[0] = wp[nt * 16 + lo]; b[1] = wp[64 + nt * 16 + lo];
      acc[nt] = wmma4(a, b, acc[nt]);
    }
  }
#pragma unroll
  for (int nt = 0; nt < 4; ++nt) {
    const float bb = bv[nt * 16 + lo];
#pragma unroll
    for (int j = 0; j < 8; ++j)
      tl[(j + 8 * hi) * 68 + nt * 16 + lo] = acc[nt][j] + bb;
  }
  __syncthreads();

  v8f h[4] = {z, z, z, z};
  for (int k0 = 0; k0 < 64; k0 += 4) {               // K rows [0,64): self half
    const int kb = k0 + 2 * hi;
    v2f a = *(const v2f*)&sl[lo * 68 + kb];
    const float* wp = Ww + (size_t)kb * 64;
#pragma unroll
    for (int nt = 0; nt < 4; ++nt) {
      v2f b; b[0] = wp[nt * 16 + lo]; b[1] = wp[64 + nt * 16 + lo];
      h[nt] = wmma4(a, b, h[nt]);
    }
  }
  for (int k0 = 0; k0 < 64; k0 += 4) {               // K rows [64,128): agg half
    const int kb = k0 + 2 * hi;
    v2f a = *(const v2f*)&tl[lo * 68 + kb];
    const float* wp = Ww + (size_t)(64 + kb) * 64;
#pragma unroll
    for (int nt = 0; nt < 4; ++nt) {
      v2f b; b[0] = wp[nt * 16 + lo]; b[1] = wp[64 + nt * 16 + lo];
      h[nt] = wmma4(a, b, h[nt]);
    }
  }
#pragma unroll
  for (int nt = 0; nt < 4; ++nt) {
    const float bb = bw[nt * 16 + lo];
#pragma unroll
    for (int j = 0; j < 8; ++j) {
      const int m = j + 8 * hi;
      float v = h[nt][j] + bb;
      if (do_relu) v = fmaxf(v, 0.f);
      out[(size_t)(row0 + m) * 64 + nt * 16 + lo] = v;
    }
  }
}

// ---------------------------------------------------------------------------
extern "C" void kernel_launch(void* const* d_in, const int* in_sizes, int n_in,
                              void* d_out, int out_size, void* d_ws, size_t ws_size,
                              hipStream_t stream)
{
  (void)in_sizes; (void)n_in; (void)out_size; (void)ws_size;
  const int*   seeds   = (const int*)d_in[0];
  const int*   neigh1  = (const int*)d_in[1];
  const int*   neigh2  = (const int*)d_in[2];
  const int*   ufeat   = (const int*)d_in[3];
  const int*   ifeat   = (const int*)d_in[4];
  const int*   utext   = (const int*)d_in[5];
  const int*   itext   = (const int*)d_in[6];
  const float* uid     = (const float*)d_in[7];
  const float* iid     = (const float*)d_in[8];
  const float* ufemb   = (const float*)d_in[9];
  const float* ifemb   = (const float*)d_in[10];
  const float* wemb    = (const float*)d_in[11];
  const float* uw300   = (const float*)d_in[12];
  const float* iw300   = (const float*)d_in[13];
  const float* is768   = (const float*)d_in[14];
  const float* unum    = (const float*)d_in[15];
  const float* inum    = (const float*)d_in[16];
  const float* Wnum_u  = (const float*)d_in[17];
  const float* bnum_u  = (const float*)d_in[18];
  const float* Wnum_i  = (const float*)d_in[19];
  const float* bnum_i  = (const float*)d_in[20];
  const float* Wproj_u = (const float*)d_in[21];
  const float* bproj_u = (const float*)d_in[22];
  const float* Wproj_i = (const float*)d_in[23];
  const float* bproj_i = (const float*)d_in[24];
  const float* W_w     = (const float*)d_in[25];
  const float* b_w     = (const float*)d_in[26];
  const float* W_v     = (const float*)d_in[27];
  const float* b_v     = (const float*)d_in[28];

  float* ws  = (float*)d_ws;
  float* x   = ws;                                    // [150000,64]
  float* x1g = x   + (size_t)(N_USER + M_ITEM) * 64;  // [40960,64]
  float* x0g = x1g + (size_t)BATCH * SNB * 64;        // [2048,64]
  float* m2  = x0g + (size_t)BATCH * 64;              // [40960,64]
  float* m1  = m2  + (size_t)BATCH * SNB * 64;        // [2048,64]
  float* h1  = m1  + (size_t)BATCH * 64;              // [40960,64]
  float* h0  = h1  + (size_t)BATCH * SNB * 64;        // [2048,64]
  float* mh1 = h0  + (size_t)BATCH * 64;              // [2048,64]

  // ---- stage 1: node embedding table ----
  embed_user_kernel<<<N_USER / 16, 64, 0, stream>>>(
      ufeat, utext, uid, ufemb, wemb, uw300, unum,
      Wnum_u, bnum_u, Wproj_u, bproj_u, x);
  embed_item_kernel<<<M_ITEM / 16, 64, 0, stream>>>(
      ifeat, itext, iid, ifemb, wemb, iw300, is768, inum,
      Wnum_i, bnum_i, Wproj_i, bproj_i, x + (size_t)N_USER * 64);

  // ---- stage 2: gathers + group means (x table is L2-resident) ----
  const int BS = BATCH * SNB;
  mean_gather_kernel<<<BS / 8,    256, 0, stream>>>(x, neigh2, m2,  BS,    SNB, 1.f / SNB);
  mean_gather_kernel<<<BS / 8,    256, 0, stream>>>(x, neigh1, x1g, BS,    1,   1.f);
  mean_gather_kernel<<<BATCH / 8, 256, 0, stream>>>(x, seeds,  x0g, BATCH, 1,   1.f);
  mean_gather_kernel<<<BATCH / 8, 256, 0, stream>>>(x, neigh1, m1,  BATCH, SNB, 1.f / SNB);

  const float* Wv0 = W_v;            const float* Wv1 = W_v + 64 * 64;
  const float* bv0 = b_v;            const float* bv1 = b_v + 64;
  const float* Ww0 = W_w;            const float* Ww1 = W_w + 128 * 64;
  const float* bw0 = b_w;            const float* bw1 = b_w + 64;

  // ---- stage 3: fused SAGE layers ----
  sage_kernel<<<BS / 16,    32, 0, stream>>>(x1g, m2, Wv0, bv0, Ww0, bw0, h1, 1);
  sage_kernel<<<BATCH / 16, 32, 0, stream>>>(x0g, m1, Wv0, bv0, Ww0, bw0, h0, 1);
  mean_gather_kernel<<<BATCH / 8, 256, 0, stream>>>(h1, nullptr, mh1, BATCH, SNB, 1.f / SNB);
  sage_kernel<<<BATCH / 16, 32, 0, stream>>>(h0, mh1, Wv1, bv1, Ww1, bw1,
                                             (float*)d_out, 0);
}